// NeuralMemory_16827681866251
// MI455X (gfx1250) — compile-verified
//
#include <hip/hip_runtime.h>
#include <hip/hip_bf16.h>

typedef __attribute__((ext_vector_type(2))) float v2f;
typedef __attribute__((ext_vector_type(8))) float v8f;

namespace {

constexpr int B_ = 4, T_ = 64, C_ = 64, D_ = 256, DE_ = 512;
constexpr int R_ = B_ * T_ * C_;                       // 16384 rows of x/q/k/v
constexpr long long WSLOT = (long long)D_ * DE_;       // 131072 elems per weight
constexpr long long PB = 4 * WSLOT;                    // per-batch param block
constexpr long long PS = (long long)B_ * PB;           // all batches
constexpr float MAX_LR_ = 0.1f;

// LDS panel: 32(K) x 64(N) staged per chunk, padded rows to kill bank
// conflicts and keep 16B alignment for async b128 stores.
constexpr int KC_ = 32;
constexpr int PANEL_ = 2304;   // floats per buffer (32*72 == 64*36)
constexpr int SMEM_BYTES_ = 2 * PANEL_ * 4;

__device__ __forceinline__ float sigm(float x) { return 1.f / (1.f + expf(-x)); }

// jax.nn.gelu default: tanh approximation
__device__ __forceinline__ float gelu_f(float x) {
  float u = 0.7978845608028654f * (x + 0.044715f * x * x * x);
  return 0.5f * x * (1.f + tanhf(u));
}
__device__ __forceinline__ float dgelu_f(float x) {
  float u = 0.7978845608028654f * (x + 0.044715f * x * x * x);
  float t = tanhf(u);
  float du = 0.7978845608028654f * (1.f + 3.f * 0.044715f * x * x);
  return 0.5f * (1.f + t) + 0.5f * x * (1.f - t * t) * du;
}

enum { EPI_NONE = 0, EPI_GELU_BOTH, EPI_GELU_ONLY, EPI_ADD, EPI_DH2, EPI_DGELU };

// Batched WMMA-f32 GEMM: C = op(A) @ op(B) (+ fused epilogue).
// One wave -> one 16x64 strip (4 f32 WMMA accumulators); all WAVES waves of a
// block share one 64-wide N strip, whose B panel is double-buffer staged into
// LDS via global_load_async_to_lds_b128 (ASYNCcnt-ordered).
// TA: A stored [K x M] (use A^T).  TB: B stored [N x K] (use B^T).
template <int TA, int TB, int WAVES>
__global__ __launch_bounds__(WAVES * 32) void wmma_gemm_k(
    const float* __restrict__ A, const float* __restrict__ Bm,
    float* __restrict__ Cm, int M, int N, int K, int lda, int ldb, int ldc,
    long long bsA, long long bsB, long long bsC, int epi,
    const float* __restrict__ E0, int lde0, long long bse0,
    const float* __restrict__ E1, int lde1, long long bse1,
    float* __restrict__ O2, int ldo2, long long bso2, float scale) {
  extern __shared__ float smem[];
  constexpr int NTH = WAVES * 32;
  constexpr int NI = (KC_ * 64) / (4 * NTH);  // async issues per thread/chunk

  const int wave = threadIdx.x >> 5;
  const int lane = threadIdx.x & 31;
  const int mBlocks = M / (16 * WAVES);
  const int mb = blockIdx.x % mBlocks;
  const int ns = blockIdx.x / mBlocks;
  const int tm = mb * WAVES + wave;
  const int n0 = ns << 6;

  const long long b = blockIdx.y;
  A += b * bsA;
  Bm += b * bsB;
  Cm += b * bsC;
  const float* e0 = E0 ? E0 + b * bse0 : nullptr;
  const float* e1 = E1 ? E1 + b * bse1 : nullptr;
  float* o2 = O2 ? O2 + b * bso2 : nullptr;

  // ISA 7.12.2 f32 fragment mapping (wave32):
  //  A 16x4: M = lane&15; lanes 0-15 hold K={0,1}, lanes 16-31 K={2,3}.
  //  C/D 16x16: row = r + (lane>=16)*8, col = lane&15.
  const int mrow = (tm << 4) + (lane & 15);
  const int nl = lane & 15;
  const int khalf = (lane >> 4) << 1;

  // Cooperative async stage of one K-chunk of the B panel into LDS.
  auto stage = [&](int bufsel, int k0) {
    const int tid = threadIdx.x;
#pragma unroll
    for (int ee = tid * 4; ee < KC_ * 64; ee += NTH * 4) {
      const float* src;
      int off;
      if constexpr (TB) {  // B stored [N x K]; LDS layout [nn][36]
        int nn = ee >> 5, kc = ee & 31;
        src = Bm + (long long)(n0 + nn) * ldb + (k0 + kc);
        off = (bufsel * PANEL_ + nn * 36 + kc) * 4;
      } else {             // B stored [K x N]; LDS layout [kc][72]
        int kc = ee >> 6, nn = ee & 63;
        src = Bm + (long long)(k0 + kc) * ldb + (n0 + nn);
        off = (bufsel * PANEL_ + kc * 72 + nn) * 4;
      }
      asm volatile("global_load_async_to_lds_b128 %0, %1, off" ::"v"(off),
                   "v"(src)
                   : "memory");
    }
  };

  v8f acc0 = {}, acc1 = {}, acc2 = {}, acc3 = {};
  int bufsel = 0;
  stage(0, 0);
  for (int k0 = 0; k0 < K; k0 += KC_) {
    const bool more = (k0 + KC_) < K;
    if (more) stage(bufsel ^ 1, k0 + KC_);
    // ASYNC loads complete in order: waiting down to NI (the next chunk's
    // in-flight issues) guarantees the current chunk has landed in LDS.
    if (more)
      asm volatile("s_wait_asynccnt %0" ::"i"(NI) : "memory");
    else
      asm volatile("s_wait_asynccnt 0" ::: "memory");
    __syncthreads();
    const float* sb = smem + bufsel * PANEL_;
#pragma unroll
    for (int kk = 0; kk < KC_; kk += 4) {
      const int kb = kk + khalf;   // local k (this half-wave's pair)
      const int kg = k0 + kb;      // global k
      v2f a;
      if constexpr (TA) {
        a.x = A[(long long)(kg)*lda + mrow];
        a.y = A[(long long)(kg + 1) * lda + mrow];
      } else {
        const float2 t = *(const float2*)(A + (long long)mrow * lda + kg);
        a.x = t.x;
        a.y = t.y;
      }
      v2f b0, b1, b2, b3;
      if constexpr (TB) {
        const float2 t0 = *(const float2*)(sb + (nl)*36 + kb);
        const float2 t1 = *(const float2*)(sb + (nl + 16) * 36 + kb);
        const float2 t2 = *(const float2*)(sb + (nl + 32) * 36 + kb);
        const float2 t3 = *(const float2*)(sb + (nl + 48) * 36 + kb);
        b0.x = t0.x; b0.y = t0.y; b1.x = t1.x; b1.y = t1.y;
        b2.x = t2.x; b2.y = t2.y; b3.x = t3.x; b3.y = t3.y;
      } else {
        const float* r0 = sb + kb * 72;
        const float* r1 = sb + (kb + 1) * 72;
        b0.x = r0[nl];      b0.y = r1[nl];
        b1.x = r0[nl + 16]; b1.y = r1[nl + 16];
        b2.x = r0[nl + 32]; b2.y = r1[nl + 32];
        b3.x = r0[nl + 48]; b3.y = r1[nl + 48];
      }
      acc0 = __builtin_amdgcn_wmma_f32_16x16x4_f32(false, a, false, b0, (short)0, acc0, false, false);
      acc1 = __builtin_amdgcn_wmma_f32_16x16x4_f32(false, a, false, b1, (short)0, acc1, false, false);
      acc2 = __builtin_amdgcn_wmma_f32_16x16x4_f32(false, a, false, b2, (short)0, acc2, false, false);
      acc3 = __builtin_amdgcn_wmma_f32_16x16x4_f32(false, a, false, b3, (short)0, acc3, false, false);
    }
    __syncthreads();
    bufsel ^= 1;
  }

  v8f accs[4] = {acc0, acc1, acc2, acc3};
  const int rbase = (tm << 4) + ((lane >> 4) << 3);
#pragma unroll
  for (int s = 0; s < 4; ++s) {
    const int col = n0 + nl + s * 16;
#pragma unroll
    for (int r = 0; r < 8; ++r) {
      const int row = rbase + r;
      const long long idx = (long long)row * ldc + col;
      const float val = accs[s][r];
      if (epi == EPI_NONE) {
        Cm[idx] = val;
      } else if (epi == EPI_GELU_BOTH) {
        o2[(long long)row * ldo2 + col] = val;
        Cm[idx] = gelu_f(val);
      } else if (epi == EPI_GELU_ONLY) {
        Cm[idx] = gelu_f(val);
      } else if (epi == EPI_ADD) {
        Cm[idx] = val + e0[(long long)row * lde0 + col];
      } else if (epi == EPI_DH2) {
        Cm[idx] = scale * (val + e0[(long long)row * lde0 + col] -
                           e1[(long long)row * lde1 + col]);
      } else {  // EPI_DGELU
        Cm[idx] = val * dgelu_f(e0[(long long)row * lde0 + col]);
      }
    }
  }
}

// Row-wise l2 normalize (ncols == blockDim.x == 256)
__global__ __launch_bounds__(256) void l2norm_rows_k(float* __restrict__ p, int ncols) {
  float* pr = p + (long long)blockIdx.x * ncols;
  float v = pr[threadIdx.x];
  float ss = v * v;
  for (int off = 16; off; off >>= 1) ss += __shfl_down(ss, off, 32);
  __shared__ float red[8];
  __shared__ float inv;
  if ((threadIdx.x & 31) == 0) red[threadIdx.x >> 5] = ss;
  __syncthreads();
  if (threadIdx.x == 0) {
    float s = 0.f;
    for (int i = 0; i < 8; ++i) s += red[i];
    inv = 1.f / fmaxf(sqrtf(s), 1e-12f);
  }
  __syncthreads();
  pr[threadIdx.x] = v * inv;
}

// Per-(b,t) gate scalars: mean over chunk of sigmoid(x_t @ w)
__global__ __launch_bounds__(C_) void gates_k(const float* __restrict__ x,
                                              const float* __restrict__ wlr,
                                              const float* __restrict__ wf,
                                              const float* __restrict__ we,
                                              float* __restrict__ gates) {
  const int bt = blockIdx.x;  // b*T + t
  const int c = threadIdx.x;
  const float* row = x + ((long long)bt * C_ + c) * D_;
  float d0 = 0.f, d1 = 0.f, d2 = 0.f;
  for (int i = 0; i < D_; ++i) {
    float v = row[i];
    d0 += v * wlr[i];
    d1 += v * wf[i];
    d2 += v * we[i];
  }
  __shared__ float s0[C_], s1[C_], s2[C_];
  s0[c] = sigm(d0);
  s1[c] = sigm(d1);
  s2[c] = sigm(d2);
  __syncthreads();
  if (c == 0) {
    float a = 0.f, bb = 0.f, cc = 0.f;
    for (int i = 0; i < C_; ++i) { a += s0[i]; bb += s1[i]; cc += s2[i]; }
    gates[bt] = MAX_LR_ * (a / C_);
    gates[B_ * T_ + bt] = bb / C_;
    gates[2 * B_ * T_ + bt] = cc / C_;
  }
}

__global__ __launch_bounds__(256) void init_params_k(
    const float* __restrict__ w10, const float* __restrict__ w20,
    const float* __restrict__ w11, const float* __restrict__ w21,
    float* __restrict__ params, float* __restrict__ mom) {
  long long idx = (long long)blockIdx.x * blockDim.x + threadIdx.x;
  if (idx >= PS) return;
  long long j = idx % PB;
  int slot = (int)(j / WSLOT);
  long long r = j % WSLOT;
  const float* src = slot == 0 ? w10 : slot == 1 ? w20 : slot == 2 ? w11 : w21;
  params[idx] = src[r];
  mom[idx] = 0.f;
}

__global__ __launch_bounds__(256) void update_k(float* __restrict__ params,
                                                float* __restrict__ mom,
                                                const float* __restrict__ grad,
                                                const float* __restrict__ gates,
                                                int t) {
  long long idx = (long long)blockIdx.x * blockDim.x + threadIdx.x;
  if (idx >= PS) return;
  int b = (int)(idx / PB);
  int bt = b * T_ + t;
  float lr = gates[bt];
  float forget = gates[B_ * T_ + bt];
  float eta = gates[2 * B_ * T_ + bt];
  float m = eta * mom[idx] - lr * grad[idx];
  mom[idx] = m;
  params[idx] = params[idx] * (1.f - forget) + m;
}

__global__ __launch_bounds__(256) void mean_params_k(const float* __restrict__ params,
                                                     float* __restrict__ fin) {
  long long idx = (long long)blockIdx.x * blockDim.x + threadIdx.x;
  if (idx >= PB) return;
  float s = 0.f;
  for (int b = 0; b < B_; ++b) s += params[b * PB + idx];
  fin[idx] = s * (1.f / B_);
}

static void launch_gemm(hipStream_t st, int TA, int TB, const float* A,
                        const float* Bm, float* Cm, int M, int N, int K,
                        int lda, int ldb, int ldc, long long bsA, long long bsB,
                        long long bsC, int batch, int epi, const float* E0,
                        int lde0, long long bse0, const float* E1, int lde1,
                        long long bse1, float* O2, int ldo2, long long bso2,
                        float scale) {
  const int waves = (M % 128 == 0) ? 8 : 4;  // exact cover: M in {64,256,512,16384}
  dim3 g((unsigned)((N >> 6) * (M / (16 * waves))), (unsigned)batch);
  dim3 blk((unsigned)(waves * 32));
#define ARGS_ A, Bm, Cm, M, N, K, lda, ldb, ldc, bsA, bsB, bsC, epi, E0, lde0, \
              bse0, E1, lde1, bse1, O2, ldo2, bso2, scale
  if (!TA && !TB) {
    if (waves == 8) wmma_gemm_k<0, 0, 8><<<g, blk, SMEM_BYTES_, st>>>(ARGS_);
    else            wmma_gemm_k<0, 0, 4><<<g, blk, SMEM_BYTES_, st>>>(ARGS_);
  } else if (!TA && TB) {
    if (waves == 8) wmma_gemm_k<0, 1, 8><<<g, blk, SMEM_BYTES_, st>>>(ARGS_);
    else            wmma_gemm_k<0, 1, 4><<<g, blk, SMEM_BYTES_, st>>>(ARGS_);
  } else {
    if (waves == 8) wmma_gemm_k<1, 0, 8><<<g, blk, SMEM_BYTES_, st>>>(ARGS_);
    else            wmma_gemm_k<1, 0, 4><<<g, blk, SMEM_BYTES_, st>>>(ARGS_);
  }
#undef ARGS_
}

}  // namespace

extern "C" void kernel_launch(void* const* d_in, const int* in_sizes, int n_in,
                              void* d_out, int out_size, void* d_ws,
                              size_t ws_size, hipStream_t stream) {
  (void)in_sizes; (void)n_in; (void)out_size; (void)ws_size;
  const float* x = (const float*)d_in[0];
  const float* Wq = (const float*)d_in[1];
  const float* Wk = (const float*)d_in[2];
  const float* Wv = (const float*)d_in[3];
  const float* wlr = (const float*)d_in[4];
  const float* wf = (const float*)d_in[5];
  const float* we = (const float*)d_in[6];
  const float* w10 = (const float*)d_in[7];
  const float* w20 = (const float*)d_in[8];
  const float* w11 = (const float*)d_in[9];
  const float* w21 = (const float*)d_in[10];

  float* ws = (float*)d_ws;
  const long long RD = (long long)R_ * D_;
  float* q = ws;
  float* kbuf = q + RD;
  float* vbuf = kbuf + RD;
  float* gates = vbuf + RD;       // 3*B*T floats
  float* params = gates + 1024;
  float* mom = params + PS;
  float* grad = mom + PS;
  float* fin = grad + PS;         // PB floats
  float* u1 = fin + PB;
  float* a1 = u1 + (long long)B_ * C_ * DE_;
  float* h1 = a1 + (long long)B_ * C_ * DE_;
  float* u2 = h1 + (long long)B_ * C_ * D_;
  float* a2 = u2 + (long long)B_ * C_ * DE_;
  float* dh2 = a2 + (long long)B_ * C_ * DE_;
  float* du2 = dh2 + (long long)B_ * C_ * D_;
  float* dh1 = du2 + (long long)B_ * C_ * DE_;
  float* du1 = dh1 + (long long)B_ * C_ * D_;

  const long long sCD = (long long)C_ * D_;    // step activation stride [C x D]
  const long long sCE = (long long)C_ * DE_;   // step activation stride [C x DE]
  const long long bsK = (long long)T_ * C_ * D_;  // batch stride inside k/v/x

  // --- Projections q/k/v: [16384x256] x [256x256] ---
  launch_gemm(stream, 0, 0, x, Wq, q, R_, D_, D_, D_, D_, D_, 0, 0, 0, 1,
              EPI_NONE, nullptr, 0, 0, nullptr, 0, 0, nullptr, 0, 0, 0.f);
  launch_gemm(stream, 0, 0, x, Wk, kbuf, R_, D_, D_, D_, D_, D_, 0, 0, 0, 1,
              EPI_NONE, nullptr, 0, 0, nullptr, 0, 0, nullptr, 0, 0, 0.f);
  launch_gemm(stream, 0, 0, x, Wv, vbuf, R_, D_, D_, D_, D_, D_, 0, 0, 0, 1,
              EPI_NONE, nullptr, 0, 0, nullptr, 0, 0, nullptr, 0, 0, 0.f);
  l2norm_rows_k<<<R_, 256, 0, stream>>>(q, D_);
  l2norm_rows_k<<<R_, 256, 0, stream>>>(kbuf, D_);
  gates_k<<<B_ * T_, C_, 0, stream>>>(x, wlr, wf, we, gates);
  init_params_k<<<(int)((PS + 255) / 256), 256, 0, stream>>>(w10, w20, w11, w21,
                                                             params, mom);

  const float dloss = 2.f / (float)(C_ * D_);
  // --- Sequential scan over T ---
  for (int t = 0; t < T_; ++t) {
    const float* kt = kbuf + (long long)t * C_ * D_;
    const float* vt = vbuf + (long long)t * C_ * D_;
    const float* W10p = params;
    const float* W20p = params + WSLOT;
    const float* W11p = params + 2 * WSLOT;
    const float* W21p = params + 3 * WSLOT;

    // u1 = kt @ W1_0 ; a1 = gelu(u1)
    launch_gemm(stream, 0, 0, kt, W10p, a1, C_, DE_, D_, D_, DE_, DE_, bsK, PB,
                sCE, B_, EPI_GELU_BOTH, nullptr, 0, 0, nullptr, 0, 0, u1, DE_,
                sCE, 0.f);
    // h1 = kt + a1 @ W2_0
    launch_gemm(stream, 0, 0, a1, W20p, h1, C_, D_, DE_, DE_, D_, D_, sCE, PB,
                sCD, B_, EPI_ADD, kt, D_, bsK, nullptr, 0, 0, nullptr, 0, 0, 0.f);
    // u2 = h1 @ W1_1 ; a2 = gelu(u2)
    launch_gemm(stream, 0, 0, h1, W11p, a2, C_, DE_, D_, D_, DE_, DE_, sCD, PB,
                sCE, B_, EPI_GELU_BOTH, nullptr, 0, 0, nullptr, 0, 0, u2, DE_,
                sCE, 0.f);
    // dh2 = (2/(C*D)) * (h1 + a2 @ W2_1 - vt)
    launch_gemm(stream, 0, 0, a2, W21p, dh2, C_, D_, DE_, DE_, D_, D_, sCE, PB,
                sCD, B_, EPI_DH2, h1, D_, sCD, vt, D_, bsK, nullptr, 0, 0, dloss);
    // du2 = (dh2 @ W2_1^T) * gelu'(u2)
    launch_gemm(stream, 0, 1, dh2, W21p, du2, C_, DE_, D_, D_, D_, DE_, sCD, PB,
                sCE, B_, EPI_DGELU, u2, DE_, sCE, nullptr, 0, 0, nullptr, 0, 0, 0.f);
    // gW1_1 = h1^T @ du2
    launch_gemm(stream, 1, 0, h1, du2, grad + 2 * WSLOT, D_, DE_, C_, D_, DE_,
                DE_, sCD, sCE, PB, B_, EPI_NONE, nullptr, 0, 0, nullptr, 0, 0,
                nullptr, 0, 0, 0.f);
    // dh1 = dh2 + du2 @ W1_1^T
    launch_gemm(stream, 0, 1, du2, W11p, dh1, C_, D_, DE_, DE_, DE_, D_, sCE,
                PB, sCD, B_, EPI_ADD, dh2, D_, sCD, nullptr, 0, 0, nullptr, 0, 0,
                0.f);
    // gW2_1 = a2^T @ dh2
    launch_gemm(stream, 1, 0, a2, dh2, grad + 3 * WSLOT, DE_, D_, C_, DE_, D_,
                D_, sCE, sCD, PB, B_, EPI_NONE, nullptr, 0, 0, nullptr, 0, 0,
                nullptr, 0, 0, 0.f);
    // gW2_0 = a1^T @ dh1
    launch_gemm(stream, 1, 0, a1, dh1, grad + WSLOT, DE_, D_, C_, DE_, D_, D_,
                sCE, sCD, PB, B_, EPI_NONE, nullptr, 0, 0, nullptr, 0, 0,
                nullptr, 0, 0, 0.f);
    // du1 = (dh1 @ W2_0^T) * gelu'(u1)
    launch_gemm(stream, 0, 1, dh1, W20p, du1, C_, DE_, D_, D_, D_, DE_, sCD, PB,
                sCE, B_, EPI_DGELU, u1, DE_, sCE, nullptr, 0, 0, nullptr, 0, 0,
                0.f);
    // gW1_0 = kt^T @ du1
    launch_gemm(stream, 1, 0, kt, du1, grad, D_, DE_, C_, D_, DE_, DE_, bsK,
                sCE, PB, B_, EPI_NONE, nullptr, 0, 0, nullptr, 0, 0, nullptr, 0,
                0, 0.f);
    // momentum + forget-gate update
    update_k<<<(int)((PS + 255) / 256), 256, 0, stream>>>(params, mom, grad,
                                                          gates, t);
  }

  // --- Mean over batch, then final MLP applied to q ---
  mean_params_k<<<(int)((PB + 255) / 256), 256, 0, stream>>>(params, fin);

  float* a1f = kbuf;  // reuse k+v region: R*DE floats
  float* h1f = mom;   // reuse mom+grad region: R*D floats
  float* a2f = kbuf;  // reuse again after a1f consumed
  float* out = (float*)d_out;

  // a1f = gelu(q @ finW1_0)
  launch_gemm(stream, 0, 0, q, fin, a1f, R_, DE_, D_, D_, DE_, DE_, 0, 0, 0, 1,
              EPI_GELU_ONLY, nullptr, 0, 0, nullptr, 0, 0, nullptr, 0, 0, 0.f);
  // h1f = q + a1f @ finW2_0
  launch_gemm(stream, 0, 0, a1f, fin + WSLOT, h1f, R_, D_, DE_, DE_, D_, D_, 0,
              0, 0, 1, EPI_ADD, q, D_, 0, nullptr, 0, 0, nullptr, 0, 0, 0.f);
  // a2f = gelu(h1f @ finW1_1)
  launch_gemm(stream, 0, 0, h1f, fin + 2 * WSLOT, a2f, R_, DE_, D_, D_, DE_,
              DE_, 0, 0, 0, 1, EPI_GELU_ONLY, nullptr, 0, 0, nullptr, 0, 0,
              nullptr, 0, 0, 0.f);
  // out = h1f + a2f @ finW2_1
  launch_gemm(stream, 0, 0, a2f, fin + 3 * WSLOT, out, R_, D_, DE_, DE_, D_, D_,
              0, 0, 0, 1, EPI_ADD, h1f, D_, 0, nullptr, 0, 0, nullptr, 0, 0,
              0.f);
}